// S4DKernel_781684048340
// MI455X (gfx1250) — compile-verified
//
#include <hip/hip_runtime.h>

// S4D log-Vandermonde kernel for MI455X (gfx1250), wave32.
//
// K[h, a+16b] = Re{ Σ_n (2*Cm[h,n]*w[h,n]^a) * (w[h,n]^16)^b },  w = exp(dt*A)
//             = Σ_n Gr[a,n]*Pr[n,b] + (-Gi[a,n])*Pi[n,b]
// i.e. per h: two real 16x128x64 GEMMs -> V_WMMA_F32_16X16X4_F32.
// G/P built in LDS via complex-multiply recurrences (no per-element exp).

typedef __attribute__((ext_vector_type(2))) float v2f;
typedef __attribute__((ext_vector_type(8))) float v8f;

#define HH 1024
#define NN 64
#define LL 2048
#define GP 68      // G row pitch (floats): lane stride 68 -> bank stride 4, conflict-free
#define PP 66      // P row pitch (floats): write stride 66 -> bank stride 2, conflict-free

__global__ __launch_bounds__(128) void s4d_wmma_kernel(
    const float* __restrict__ A_real, const float* __restrict__ A_imag,
    const float* __restrict__ Bm,     const float* __restrict__ Cm,
    const float* __restrict__ inv_dt, float* __restrict__ out)
{
  __shared__ float Gr [16 * GP];   // (2*Cmod*w^a).real,  a rows, n cols
  __shared__ float Gmi[16 * GP];   // -(2*Cmod*w^a).imag  (pre-negated: f32 WMMA has no A/B neg)
  __shared__ float Pr [NN * PP];   // (w16 powers).real, 64-column slab per pass
  __shared__ float Pi [NN * PP];
  __shared__ float w16r[NN], w16i[NN];

  const int h   = blockIdx.x;
  const int tid = threadIdx.x;

  // ---------------- Phase 1: per-n setup + G matrix (threads 0..63) ----------
  if (tid < NN) {
    const int n = tid;
    const float dt = __expf(inv_dt[h]);
    const float Ar = -__expf(A_real[h * NN + n]);
    const float Ai = A_imag[h * NN + n];
    const float dr = Ar * dt, di = Ai * dt;
    // w = exp(dtA)
    const float ew = __expf(dr);
    float s, c;
    __sincosf(di, &s, &c);
    const float wr = ew * c, wi = ew * s;
    // Bc * Cc
    const float Br = Bm[(h * NN + n) * 2 + 0], Bi = Bm[(h * NN + n) * 2 + 1];
    const float Cr = Cm[(h * NN + n) * 2 + 0], Ci = Cm[(h * NN + n) * 2 + 1];
    const float bcr = Br * Cr - Bi * Ci;
    const float bci = Br * Ci + Bi * Cr;
    // (w - 1) / A
    const float nr  = wr - 1.0f, ni = wi;
    const float inv = 1.0f / (Ar * Ar + Ai * Ai);
    const float qr  = (nr * Ar + ni * Ai) * inv;
    const float qi  = (ni * Ar - nr * Ai) * inv;
    // g = 2 * (Bc*Cc) * (w-1)/A ; then G[a,n] = g * w^a
    float gr = 2.0f * (bcr * qr - bci * qi);
    float gi = 2.0f * (bcr * qi + bci * qr);
    #pragma unroll
    for (int a = 0; a < 16; ++a) {
      Gr [a * GP + n] =  gr;
      Gmi[a * GP + n] = -gi;
      const float t = gr * wr - gi * wi;
      gi = gr * wi + gi * wr;
      gr = t;
    }
    // w16 = w^16 (4 squarings)
    float xr = wr, xi = wi;
    #pragma unroll
    for (int sq = 0; sq < 4; ++sq) { const float t = xr * xr - xi * xi; xi = 2.0f * xr * xi; xr = t; }
    w16r[n] = xr; w16i[n] = xi;
  }
  __syncthreads();

  const int wave = tid >> 5;
  const int lane = tid & 31;
  const int lh   = lane >> 4;      // 0: lanes 0-15, 1: lanes 16-31
  const int l15  = lane & 15;
  float* const o = out + (size_t)h * LL;

  // Two passes over b in 64-column slabs keeps static LDS ~43 KB.
  for (int pass = 0; pass < 2; ++pass) {
    // -------------- Phase 2: build P slab: P[n][half*32 + j] = w16^(4*pass + 2*half + j)
    {
      const int n    = tid & 63;
      const int half = tid >> 6;           // 32 columns per thread
      const float sr = w16r[n], si = w16i[n];
      const float w32r = sr * sr - si * si;
      const float w32i = 2.0f * sr * si;
      // start = w32^(2*pass + half)
      float cr = 1.0f, ci = 0.0f;
      const int m = 2 * pass + half;
      for (int i = 0; i < m; ++i) { const float t = cr * w32r - ci * w32i; ci = cr * w32i + ci * w32r; cr = t; }
      const int base = n * PP + half * 32;
      for (int j = 0; j < 32; ++j) {
        Pr[base + j] = cr;
        Pi[base + j] = ci;
        const float t = cr * sr - ci * si;
        ci = cr * si + ci * sr;
        cr = t;
      }
    }
    __syncthreads();

    // -------------- Phase 3: Kt[b,a] = P^T x G^T via V_WMMA_F32_16X16X4_F32
    // A-frag (16x4, M = b-col of slab): lane<16 -> K{0,1}, lane>=16 -> K{2,3}
    // B-frag (4x16, N = a):            VGPR0 rows {0,2}, VGPR1 rows {1,3}
    {
      const int b0l = wave * 16;                 // slab-local column base (4 waves x 16 = 64)
      const int b0g = pass * 64 + b0l;           // global b base
      v8f acc = {};
      #pragma unroll
      for (int k = 0; k < NN; k += 4) {
        const int kk = k + lh * 2;
        v2f ar, ai, br, bi;
        ar.x = Pr[(kk + 0) * PP + b0l + l15];
        ar.y = Pr[(kk + 1) * PP + b0l + l15];
        ai.x = Pi[(kk + 0) * PP + b0l + l15];
        ai.y = Pi[(kk + 1) * PP + b0l + l15];
        br.x = Gr [l15 * GP + kk + 0];
        br.y = Gr [l15 * GP + kk + 1];
        bi.x = Gmi[l15 * GP + kk + 0];
        bi.y = Gmi[l15 * GP + kk + 1];
        acc = __builtin_amdgcn_wmma_f32_16x16x4_f32(false, ar, false, br,
                                                    (short)0, acc, false, false);
        acc = __builtin_amdgcn_wmma_f32_16x16x4_f32(false, ai, false, bi,
                                                    (short)0, acc, false, false);
      }
      // D: VGPR j -> row M = j + 8*lh (b within tile), col N = l15 (a).
      // l = 16*(b0g + M) + a : lanes 0-15 and 16-31 each store 64B contiguous.
      #pragma unroll
      for (int j = 0; j < 8; ++j) {
        const int bidx = j + lh * 8;
        o[16 * (b0g + bidx) + l15] = acc[j];
      }
    }
    __syncthreads();  // before next pass overwrites P
  }
}

extern "C" void kernel_launch(void* const* d_in, const int* in_sizes, int n_in,
                              void* d_out, int out_size, void* d_ws, size_t ws_size,
                              hipStream_t stream) {
  const float* A_real = (const float*)d_in[0];
  const float* A_imag = (const float*)d_in[1];
  const float* B      = (const float*)d_in[2];
  const float* C      = (const float*)d_in[3];
  const float* inv_dt = (const float*)d_in[4];
  // d_in[5] is L (== 2048), compile-time constant here.
  (void)in_sizes; (void)n_in; (void)out_size; (void)d_ws; (void)ws_size;

  s4d_wmma_kernel<<<dim3(HH), dim3(128), 0, stream>>>(
      A_real, A_imag, B, C, inv_dt, (float*)d_out);
}